// NearestNeighbor_44178033606906
// MI455X (gfx1250) — compile-verified
//
#include <hip/hip_runtime.h>

typedef __attribute__((ext_vector_type(2))) float v2f;
typedef __attribute__((ext_vector_type(8))) float v8f;

#define D_POSE     66
#define K_VEL      990    // 15 * 66
#define SAMPLE_LEN 1056   // 16 * 66
#define FUT_LEN    924    // 14 * 66
#define K_CHUNKS   248    // ceil(992/4)

// ---------------------------------------------------------------------------
// Kernel 0: reset the global argmin cell (graph-replay safe).
// ---------------------------------------------------------------------------
__global__ void nn_init_kernel(unsigned long long* best) {
    if (threadIdx.x == 0) best[0] = ~0ULL;
}

// ---------------------------------------------------------------------------
// Kernel 1: streaming distance + argmin.
//   block = 128 threads = 4 waves; each wave owns a tile of 16 samples.
//   dist[n] = sum(a^2) - sum(a * 2q)   (+||q||^2, constant, dropped)
//   cross term via V_WMMA_F32_16X16X4_F32 with B = 2q broadcast to 16 cols.
// ---------------------------------------------------------------------------
__global__ __launch_bounds__(128)
void nn_dist_kernel(const float* __restrict__ obs,
                    const float* __restrict__ train,
                    int N, unsigned long long* __restrict__ best) {
    __shared__ float qv2[992];                 // 2 * query velocity, zero padded
    __shared__ unsigned long long wbest[4];

    // Build 2*q in LDS (obs is tiny and L2-resident; cheap per block).
    for (int k = threadIdx.x; k < 992; k += blockDim.x) {
        float v = 0.0f;
        if (k < K_VEL) v = 2.0f * (obs[k + D_POSE] - obs[k]);
        qv2[k] = v;
    }
    __syncthreads();

    const int lane   = threadIdx.x & 31;
    const int wave   = threadIdx.x >> 5;
    const int sLocal = lane & 15;              // this lane's A-matrix row (sample)
    const int koff   = (lane >> 4) << 1;       // K offset within chunk: 0 or 2

    const int tileBase = blockIdx.x * 64 + wave * 16;
    int gs = tileBase + sLocal;
    int gsc = gs < N ? gs : (N - 1);           // clamp so loads stay in bounds
    const float* __restrict__ sp = train + (size_t)gsc * SAMPLE_LEN;

    v8f  c  = {0.f, 0.f, 0.f, 0.f, 0.f, 0.f, 0.f, 0.f};
    float n2 = 0.0f;

    // Hot loop: chunks 0..246 are fully in-range (k <= 986, k+1+66 <= 1053).
    for (int kk = 0; kk < K_CHUNKS - 1; ++kk) {
        const int k = kk * 4 + koff;
        if ((kk & 31) == 0) __builtin_prefetch(sp + kk * 4 + 512, 0, 3);
        v2f p0 = *(const v2f*)(sp + k);           // pose[k], pose[k+1]
        v2f p1 = *(const v2f*)(sp + k + D_POSE);  // pose[k+66], pose[k+67]
        v2f a;  a.x = p1.x - p0.x;  a.y = p1.y - p0.y;   // velocity
        v2f b = *(const v2f*)(qv2 + k);                  // 2q chunk (LDS)
        n2 += a.x * a.x + a.y * a.y;
        c = __builtin_amdgcn_wmma_f32_16x16x4_f32(
                false, a, false, b, (short)0, c, false, false);
    }

    // Tail chunk (kk = 247): k = 988 valid for koff==0, k = 990 out of range.
    {
        const int k = 988 + koff;
        v2f a; a.x = 0.0f; a.y = 0.0f;
        if (koff == 0) {
            v2f p0 = *(const v2f*)(sp + k);
            v2f p1 = *(const v2f*)(sp + k + D_POSE);
            a.x = p1.x - p0.x;  a.y = p1.y - p0.y;
        }
        v2f b = *(const v2f*)(qv2 + k);            // padded zeros beyond 989
        n2 += a.x * a.x + a.y * a.y;
        c = __builtin_amdgcn_wmma_f32_16x16x4_f32(
                false, a, false, b, (short)0, c, false, false);
    }

    // n2 partials: lane s has k%4 in {0,1}, lane s+16 has {2,3} for sample s.
    n2 += __shfl_xor(n2, 16, 32);
    // Now lane l holds full ||a||^2 of sample (l & 15).

    // D layout: c[v] = dot(sample v + 8*(lane>=16), 2q), identical across cols.
    unsigned long long key = ~0ULL;
    const int mBase = (lane >> 4) << 3;        // 0 or 8
    #pragma unroll
    for (int v = 0; v < 8; ++v) {
        const int m = v + mBase;               // tile-local sample of c[v]
        const float n2m  = __shfl(n2, m, 32);  // per-lane src -> ds_bpermute
        const float dist = n2m - c[v];
        const int   sidx = tileBase + m;
        unsigned u = __float_as_uint(dist);
        u = (u & 0x80000000u) ? ~u : (u | 0x80000000u);  // order-preserving
        unsigned long long cand =
            ((unsigned long long)u << 32) | (unsigned int)sidx;
        if (sidx >= N) cand = ~0ULL;
        key = cand < key ? cand : key;
    }

    // Wave-level min reduction.
    #pragma unroll
    for (int off = 16; off > 0; off >>= 1) {
        unsigned long long o = __shfl_xor(key, off, 32);
        key = o < key ? o : key;
    }
    if (lane == 0) wbest[wave] = key;
    __syncthreads();
    if (threadIdx.x == 0) {
        unsigned long long k0 = wbest[0];
        #pragma unroll
        for (int w = 1; w < 4; ++w) k0 = wbest[w] < k0 ? wbest[w] : k0;
        atomicMin(best, k0);
    }
}

// ---------------------------------------------------------------------------
// Kernel 2: gather the winning future pose.
// ---------------------------------------------------------------------------
__global__ void nn_gather_kernel(const unsigned long long* __restrict__ best,
                                 const float* __restrict__ fut,
                                 float* __restrict__ out) {
    const unsigned idx = (unsigned)(best[0] & 0xFFFFFFFFu);
    const float* __restrict__ src = fut + (size_t)idx * FUT_LEN;
    for (int i = threadIdx.x; i < FUT_LEN; i += blockDim.x) out[i] = src[i];
}

// ---------------------------------------------------------------------------
extern "C" void kernel_launch(void* const* d_in, const int* in_sizes, int n_in,
                              void* d_out, int out_size, void* d_ws, size_t ws_size,
                              hipStream_t stream) {
    const float* obs   = (const float*)d_in[0];   // [1, 16, 66]
    const float* train = (const float*)d_in[1];   // [N, 16, 66]
    const float* fut   = (const float*)d_in[2];   // [N, 14, 66]
    float*       out   = (float*)d_out;           // [1, 14, 66]
    unsigned long long* best = (unsigned long long*)d_ws;

    const int N = in_sizes[1] / SAMPLE_LEN;
    const int blocks = (N + 63) / 64;             // 64 samples per block

    nn_init_kernel<<<1, 32, 0, stream>>>(best);
    nn_dist_kernel<<<blocks, 128, 0, stream>>>(obs, train, N, best);
    nn_gather_kernel<<<1, 256, 0, stream>>>(best, fut, out);
}